// Visual_Affinity_Creator_Deformable_32744830665214
// MI455X (gfx1250) — compile-verified
//
#include <hip/hip_runtime.h>
#include <cmath>

// Visual affinity from deformable bilinear sampling.
// depth:  [B, 1, H, W] f32
// offset: [B, 2*K2, H, W] f32, channel c = k*2 + {0: y-off, 1: x-off}
// out:    [B, K2, H, W] f32
//
// Memory-bound: ~305 MB HBM traffic -> ~13 us floor @ 23.3 TB/s.
// Strategy: 1 thread = 4 consecutive pixels; B128 NT loads for offsets,
// B128 NT stores for output; depth gathers are regular-temporal so the
// 39 MB image stays resident in the 192 MB L2. Exact 3-D grid decomposition
// (32x8 block over x4/y, grid.z = batch) avoids all integer div/mod.

#define KK    3
#define K2C   9
#define PADC  1
#define DECAY 4.0f
#define RELT  0.3f
#define EPSV  1e-8f

typedef __attribute__((ext_vector_type(4))) float v4f;

__device__ __forceinline__ float fetch_depth(const float* __restrict__ img,
                                             int iy, int ix, int H, int W) {
  bool valid = (iy >= 0) & (iy < H) & (ix >= 0) & (ix < W);
  int iyc = min(max(iy, 0), H - 1);
  int ixc = min(max(ix, 0), W - 1);
  float v = img[(size_t)iyc * (size_t)W + (size_t)ixc];
  return valid ? v : 0.0f;
}

__device__ __forceinline__ float bilin(const float* __restrict__ img,
                                       float py, float px, int H, int W) {
  float y0f = floorf(py);
  float x0f = floorf(px);
  float wy = py - y0f;
  float wx = px - x0f;
  int y0 = (int)y0f;
  int x0 = (int)x0f;
  float v00 = fetch_depth(img, y0,     x0,     H, W);
  float v01 = fetch_depth(img, y0,     x0 + 1, H, W);
  float v10 = fetch_depth(img, y0 + 1, x0,     H, W);
  float v11 = fetch_depth(img, y0 + 1, x0 + 1, H, W);
  // Exact same product form as the reference (no lerp refactor).
  return v00 * (1.0f - wy) * (1.0f - wx)
       + v01 * (1.0f - wy) * wx
       + v10 * wy * (1.0f - wx)
       + v11 * wy * wx;
}

__global__ void __launch_bounds__(256)
deform_affinity_kernel(const float* __restrict__ depth,
                       const float* __restrict__ offset,
                       float* __restrict__ out,
                       int H, int W) {
  // block(32,8): tIdx.x -> x4 chunk, tIdx.y -> row; grid(W4/32, H/8, B)
  const int W4 = W >> 2;
  int x4 = blockIdx.x * 32 + threadIdx.x;
  int y  = blockIdx.y * 8  + threadIdx.y;
  int b  = blockIdx.z;
  if (x4 >= W4 || y >= H) return;
  int x = x4 << 2;

  const size_t HW = (size_t)H * (size_t)W;
  const float* __restrict__ dimg = depth  + (size_t)b * HW;
  const float* __restrict__ offp = offset + (size_t)b * (2 * K2C) * HW
                                          + (size_t)y * (size_t)W + (size_t)x;
  float* __restrict__ outp = out + (size_t)b * K2C * HW
                                 + (size_t)y * (size_t)W + (size_t)x;

  const float yf  = (float)y;
  const float xf0 = (float)x;

  // ---- center tap (k = 4, ky = kx = 0) ----
  v4f cy4 = __builtin_nontemporal_load((const v4f*)(offp + (size_t)8 * HW));
  v4f cx4 = __builtin_nontemporal_load((const v4f*)(offp + (size_t)9 * HW));
  float c[4];
#pragma unroll
  for (int j = 0; j < 4; ++j) {
    float py = cy4[j] + yf  + 0.0f;          // off_y + ys + ky, left-assoc
    float px = cx4[j] + (xf0 + (float)j) + 0.0f;
    c[j] = bilin(dimg, py, px, H, W);
  }

  // ---- all 9 taps ----
#pragma unroll
  for (int k = 0; k < K2C; ++k) {
    v4f r;
    if (k == K2C / 2) {
      // sampled == center exactly -> diff = 0 -> within threshold -> 1.0
      r = (v4f){1.0f, 1.0f, 1.0f, 1.0f};
    } else {
      const float kyf = (float)(k / KK - PADC);
      const float kxf = (float)(k % KK - PADC);
      v4f oy4 = __builtin_nontemporal_load((const v4f*)(offp + (size_t)(2 * k)     * HW));
      v4f ox4 = __builtin_nontemporal_load((const v4f*)(offp + (size_t)(2 * k + 1) * HW));
#pragma unroll
      for (int j = 0; j < 4; ++j) {
        float py = oy4[j] + yf + kyf;
        float px = ox4[j] + (xf0 + (float)j) + kxf;
        float s    = bilin(dimg, py, px, H, W);
        float diff = fabsf(s - c[j]);
        float rel  = diff / (c[j] + EPSV);
        float aff  = __expf(-DECAY * (diff - c[j] * RELT));
        r[j] = (rel <= RELT) ? 1.0f : aff;
      }
    }
    __builtin_nontemporal_store(r, (v4f*)(outp + (size_t)k * HW));
  }
}

extern "C" void kernel_launch(void* const* d_in, const int* in_sizes, int n_in,
                              void* d_out, int out_size, void* d_ws, size_t ws_size,
                              hipStream_t stream) {
  (void)n_in; (void)out_size; (void)d_ws; (void)ws_size;
  const float* depth  = (const float*)d_in[0];
  const float* offset = (const float*)d_in[1];
  float* out = (float*)d_out;

  const int H = 480, W = 640;
  int B = in_sizes[0] / (H * W);               // depth is [B,1,H,W]
  if (B < 1) B = 1;

  const int W4 = W / 4;                        // 160
  dim3 block(32, 8, 1);                        // 8 wave32 per workgroup
  dim3 grid((W4 + 31) / 32, (H + 7) / 8, B);   // (5, 60, 8)

  deform_affinity_kernel<<<grid, block, 0, stream>>>(depth, offset, out, H, W);
}